// NumDualDescriptorAB_2740189135096
// MI455X (gfx1250) — compile-verified
//
#include <hip/hip_runtime.h>

typedef __attribute__((ext_vector_type(2))) float v2f;
typedef __attribute__((ext_vector_type(8))) float v8f;

#define M_DIM 128
#define L_BAS 64
#define RANK  8
#define BATCH 32
#define SEQ   2048
#define NW    (SEQ - RANK + 1)                     // 2041
#define W_TILE 64
#define TILES_PER_B ((NW + W_TILE - 1) / W_TILE)   // 32
#define KCHUNK 32

// K-pair interleaved LDS layouts: element (K, x) lives at [K/2][x][K&1].
// Strides chosen so (stride % 64banks) == 32 -> lane-halves hit disjoint banks.
#define APS 160                                    // agg pair stride (>= 2*W_TILE), 160%64==32
#define MPS 288                                    // Mt pair stride (>= 2*M_DIM), 288%64==32

// Fused: windowed mean -> (64x128) @ Mt (128x128) via V_WMMA_F32_16X16X4_F32
// -> masked squared-error accumulation -> one partial sum per workgroup.
__global__ __launch_bounds__(256, 2) void ddab_fused(
    const float* __restrict__ seq, const float* __restrict__ Mmat,
    const float* __restrict__ Acoeff, const float* __restrict__ Bbasis,
    float* __restrict__ partial)
{
  __shared__ __align__(16) float aggP[(M_DIM / 2) * APS];   // 40,960 B
  __shared__ __align__(16) float mtP[(KCHUNK / 2) * MPS];   // 18,432 B
  __shared__ float redS[8];

  const int tid = threadIdx.x;
  const int b   = blockIdx.x / TILES_PER_B;
  const int wt  = blockIdx.x % TILES_PER_B;
  const int w0  = wt * W_TILE;
  const v2f* seq2 = (const v2f*)(seq + (size_t)b * (SEQ * M_DIM));

  // Phase 1: agg[w][m] = mean over RANK rows; stored K-pair interleaved.
  // w in low bits -> per-wave LDS b64 stores sweep all 64 banks (conflict-free).
  // Global 8B gathers: WG's seq footprint is ~36KB -> WGP$-resident after first touch.
  for (int i = tid; i < W_TILE * (M_DIM / 2); i += 256) {
    const int w  = i & (W_TILE - 1);
    const int mp = i >> 6;                         // 0..63 (pair of m-columns)
    v2f s = {};
#pragma unroll
    for (int r = 0; r < RANK; ++r) {
      int row = w0 + w + r;
      row = (row < SEQ) ? row : (SEQ - 1);         // clamp; masked rows dropped later
      s += seq2[row * (M_DIM / 2) + mp];
    }
    *(v2f*)&aggP[mp * APS + 2 * w] = s * (1.0f / RANK);
  }

  const int lane = tid & 31;
  const int wv   = tid >> 5;     // 8 waves
  const int rg   = wv >> 1;      // row group 0..3 -> 16 output rows each
  const int cg   = wv & 1;       // col group 0..1 -> 64 output cols each
  const int lr   = lane & 15;    // lane within half
  const int lh   = lane >> 4;    // lane half (0/1)

  v8f acc[4] = {};               // 4 tiles of 16x16 f32 per wave

  // Per-lane fragment base pointers (lane-half selects the +1 K-pair).
  const float* aBase = &aggP[lh * APS + 2 * (rg * 16 + lr)];
  const float* bBase = &mtP[lh * MPS + 2 * (cg * 64 + lr)];

  for (int k0 = 0; k0 < M_DIM; k0 += KCHUNK) {
    __syncthreads();             // aggP ready / previous mtP chunk consumed
    // Stage Mt chunk interleaved: mtP[kk/2][n][kk&1] = M[n][k0+kk].
    // n in low bits -> LDS writes stride 8B across lanes (conflict-free); global
    // reads are column gathers of the 64KB M matrix (L2/WGP$-resident).
    for (int i = tid; i < M_DIM * KCHUNK; i += 256) {
      const int kk = i >> 7;
      const int n  = i & (M_DIM - 1);
      mtP[(kk >> 1) * MPS + 2 * n + (kk & 1)] = Mmat[n * M_DIM + (k0 + kk)];
    }
    __syncthreads();

#pragma unroll
    for (int kk = 0; kk < KCHUNK; kk += 4) {
      // A fragment (16x4 f32): one aligned b64 per lane, conflict-free banks.
      const v2f a = *(const v2f*)&aBase[((k0 + kk) >> 1) * APS];
#pragma unroll
      for (int t = 0; t < 4; ++t) {
        // B fragment (4x16 f32): one aligned b64 per lane, conflict-free banks.
        const v2f bv = *(const v2f*)&bBase[(kk >> 1) * MPS + 32 * t];
        acc[t] = __builtin_amdgcn_wmma_f32_16x16x4_f32(
            false, a, false, bv, (short)0, acc[t], false, false);
      }
    }
  }

  // Epilogue: subtract Nk, square, accumulate (mask w >= NW).
  float ss = 0.0f;
#pragma unroll
  for (int t = 0; t < 4; ++t) {
    const int n = cg * 64 + t * 16 + lr;
#pragma unroll
    for (int r = 0; r < 8; ++r) {
      const int w = w0 + rg * 16 + 8 * lh + r;     // D layout: VGPR r -> M=r (+8 hi half)
      if (w < NW) {
        const int km = w & (L_BAS - 1);
        const float nk = Acoeff[n * L_BAS + km] * Bbasis[km * M_DIM + n];
        const float d  = acc[t][r] - nk;
        ss += d * d;
      }
    }
  }

  // wave32 reduce, then cross-wave via LDS
#pragma unroll
  for (int off = 16; off > 0; off >>= 1) ss += __shfl_down(ss, off);
  if (lane == 0) redS[wv] = ss;
  __syncthreads();
  if (tid == 0) {
    float tot = 0.0f;
#pragma unroll
    for (int i = 0; i < 8; ++i) tot += redS[i];
    partial[blockIdx.x] = tot;
  }
}

// Deterministic final reduction -> scalar mean
__global__ __launch_bounds__(256) void ddab_reduce(
    const float* __restrict__ partial, int n, float* __restrict__ out)
{
  __shared__ float red[256];
  float s = 0.0f;
  for (int i = threadIdx.x; i < n; i += 256) s += partial[i];
  red[threadIdx.x] = s;
  __syncthreads();
  for (int off = 128; off > 0; off >>= 1) {
    if (threadIdx.x < off) red[threadIdx.x] += red[threadIdx.x + off];
    __syncthreads();
  }
  if (threadIdx.x == 0)
    out[0] = red[0] * (1.0f / ((float)BATCH * (float)NW * (float)M_DIM));
}

extern "C" void kernel_launch(void* const* d_in, const int* in_sizes, int n_in,
                              void* d_out, int out_size, void* d_ws, size_t ws_size,
                              hipStream_t stream) {
  (void)in_sizes; (void)n_in; (void)out_size; (void)ws_size;
  const float* seq    = (const float*)d_in[0];  // (32, 2048, 128)
  const float* Mmat   = (const float*)d_in[1];  // (128, 128)
  const float* Acoeff = (const float*)d_in[2];  // (128, 64)
  const float* Bbasis = (const float*)d_in[3];  // (64, 128)
  float* out = (float*)d_out;
  float* partial = (float*)d_ws;                // 1024 floats of scratch

  const int nblocks = BATCH * TILES_PER_B;      // 1024
  ddab_fused<<<nblocks, 256, 0, stream>>>(seq, Mmat, Acoeff, Bbasis, partial);
  ddab_reduce<<<1, 256, 0, stream>>>(partial, nblocks, out);
}